// WELDON_41592463294662
// MI455X (gfx1250) — compile-verified
//
#include <hip/hip_runtime.h>

// ---------------------------------------------------------------------------
// WELDON: f = x @ W  (GEMM, fp32, V_WMMA_F32_16X16X4_F32), then per (b,e)
// top-3 + bottom-3 sum pooling over 4096 spatial positions, then L2 norm.
// x: [32, 64, 64, 512] f32  -> rows R = 131072, C = 512
// W: [512, 128] f32
// out: [32, 128] f32
//
// Data path: x staged to LDS with GLOBAL_LOAD_ASYNC_TO_LDS_B128 (ASYNCcnt),
// double-buffered so chunk kc+1 streams in while chunk kc feeds the WMMAs.
// ---------------------------------------------------------------------------

typedef __attribute__((ext_vector_type(2))) float v2f;
typedef __attribute__((ext_vector_type(8))) float v8f;

#define C_IN 512
#define E_DIM 128
#define ROWS_PER_BLOCK 128
#define KCHUNK 32
#define NCHUNK (C_IN / KCHUNK)   // 16
#define LDS_STRIDE 36            // floats per LDS row: 16B-aligned, conflict-free

#define NEG_SENT (-3.402823466e38f)
#define POS_SENT ( 3.402823466e38f)

#if __has_builtin(__builtin_amdgcn_global_load_async_to_lds_b128) && \
    __has_builtin(__builtin_amdgcn_s_wait_asynccnt)
#define USE_ASYNC 1
#else
#define USE_ASYNC 0
#endif

// exact parameter types of the async-LDS builtin (from the clang diagnostic):
//   param0: __attribute__((vector_size(16))) int  in AS(1)  (global source)
//   param1: same vector type in AS(3)                        (LDS destination)
typedef int v4i_raw __attribute__((vector_size(16)));
typedef __attribute__((address_space(1))) v4i_raw gv4i;
typedef __attribute__((address_space(3))) v4i_raw lv4i;

__device__ __forceinline__ gv4i* to_global(const void* p) {
    return (gv4i*)(unsigned long long)p;
}
__device__ __forceinline__ lv4i* to_lds(void* p) {
    // generic LDS pointer -> 32-bit LDS offset -> AS(3) pointer
    return (lv4i*)(unsigned)(unsigned long long)p;
}

__device__ __forceinline__ void ins_max3(float v, float& a, float& b, float& c) {
    // keep a >= b >= c = three largest seen
    if (v > c) {
        if (v > b) { c = b; if (v > a) { b = a; a = v; } else { b = v; } }
        else { c = v; }
    }
}
__device__ __forceinline__ void ins_min3(float v, float& a, float& b, float& c) {
    // keep a <= b <= c = three smallest seen
    if (v < c) {
        if (v < b) { c = b; if (v < a) { b = a; a = v; } else { b = v; } }
        else { c = v; }
    }
}

// ---------------------------------------------------------------------------
// Kernel 1: per block: 128 rows x 128 cols of f via WMMA fp32, then reduce
// the 128 rows to per-column top-3 / bottom-3 candidates in d_ws.
// cand layout: [b(32)][rowblk(32)][e(128)][6] floats  (max3 then min3)
// ---------------------------------------------------------------------------
__global__ __launch_bounds__(256) void weldon_gemm_pool(
    const float* __restrict__ x,
    const float* __restrict__ W,
    float* __restrict__ cand)
{
#if USE_ASYNC
    __shared__ float xs[2][ROWS_PER_BLOCK * LDS_STRIDE];
#else
    __shared__ float xs[1][ROWS_PER_BLOCK * LDS_STRIDE];
#endif

    const int rb   = blockIdx.x;          // row-block within batch (0..31)
    const int b    = blockIdx.y;          // batch (0..31)
    const int tid  = threadIdx.x;
    const int wave = tid >> 5;            // 0..7 -> owns N cols [16w, 16w+16)
    const int lane = tid & 31;
    const int laneLo = lane & 15;
    const int hi   = lane >> 4;           // 0: K pair {0,1}, 1: K pair {2,3}

    const long rowbase = (long)b * 4096 + (long)rb * ROWS_PER_BLOCK;

    // B-fragment base: lane holds W[K + hi*2 + {0,1}][wave*16 + laneLo]
    const float* wp = W + (hi * 2) * E_DIM + wave * 16 + laneLo;

    v8f acc[8];
#pragma unroll
    for (int mt = 0; mt < 8; ++mt)
#pragma unroll
        for (int r = 0; r < 8; ++r) acc[mt][r] = 0.0f;

    // per-thread staging coordinates: 4 x float4 slots of the 128x32 chunk
    int srow[4], scol[4];
#pragma unroll
    for (int i = 0; i < 4; ++i) {
        const int idx = tid + 256 * i;    // 0..1023 float4 slots
        srow[i] = idx >> 3;               // 0..127
        scol[i] = (idx & 7) * 4;          // 0,4,..,28
    }

    auto compute = [&](int kc, const float* cur) {
#pragma unroll
        for (int kk = 0; kk < KCHUNK / 4; ++kk) {
            const int k = kc * KCHUNK + kk * 4;
            // B fragment (4x16 f32): VGPR0 = row k+2*hi, VGPR1 = row k+1+2*hi
            v2f bf;
            bf.x = wp[(k + 0) * E_DIM];
            bf.y = wp[(k + 1) * E_DIM];
#pragma unroll
            for (int mt = 0; mt < 8; ++mt) {
                // A fragment (16x4 f32): lane laneLo row, K pair selected by hi
                const v2f af = *reinterpret_cast<const v2f*>(
                    &cur[(mt * 16 + laneLo) * LDS_STRIDE + kk * 4 + hi * 2]);
                acc[mt] = __builtin_amdgcn_wmma_f32_16x16x4_f32(
                    false, af, false, bf, (short)0, acc[mt], false, false);
            }
        }
    };

#if USE_ASYNC
    auto issue = [&](int kc) {
        float* dst = xs[kc & 1];
#pragma unroll
        for (int i = 0; i < 4; ++i) {
            const float* g = x + (rowbase + srow[i]) * C_IN + kc * KCHUNK + scol[i];
            float* l = &dst[srow[i] * LDS_STRIDE + scol[i]];
            __builtin_amdgcn_global_load_async_to_lds_b128(to_global(g), to_lds(l), 0, 0);
        }
    };

    issue(0);
#pragma unroll 2
    for (int kc = 0; kc < NCHUNK; ++kc) {
        if (kc + 1 < NCHUNK) {
            issue(kc + 1);
            __builtin_amdgcn_s_wait_asynccnt(4);   // own chunk-kc loads complete
        } else {
            __builtin_amdgcn_s_wait_asynccnt(0);
        }
        __syncthreads();                           // all waves' data visible
        compute(kc, xs[kc & 1]);
        __syncthreads();                           // buffer free for reuse
    }
#else
#pragma unroll 1
    for (int kc = 0; kc < NCHUNK; ++kc) {
        __syncthreads();
#pragma unroll
        for (int i = 0; i < 4; ++i) {
            const float4 v = *reinterpret_cast<const float4*>(
                x + (rowbase + srow[i]) * C_IN + kc * KCHUNK + scol[i]);
            *reinterpret_cast<float4*>(&xs[0][srow[i] * LDS_STRIDE + scol[i]]) = v;
        }
        __syncthreads();
        compute(kc, xs[0]);
    }
#endif

    // --- block-local WELDON reduction ------------------------------------
    // acc[mt][r] holds f[row = rowbase + mt*16 + r + 8*hi][col = wave*16 + laneLo]
    float a1 = NEG_SENT, a2 = NEG_SENT, a3 = NEG_SENT;
    float m1 = POS_SENT, m2 = POS_SENT, m3 = POS_SENT;
#pragma unroll
    for (int mt = 0; mt < 8; ++mt)
#pragma unroll
        for (int r = 0; r < 8; ++r) {
            const float v = acc[mt][r];
            ins_max3(v, a1, a2, a3);
            ins_min3(v, m1, m2, m3);
        }

    // merge lane pair (j, j+16): both halves hold the same column
    ins_max3(__shfl_xor(a1, 16, 32), a1, a2, a3);
    ins_max3(__shfl_xor(a2, 16, 32), a1, a2, a3);
    ins_max3(__shfl_xor(a3, 16, 32), a1, a2, a3);
    ins_min3(__shfl_xor(m1, 16, 32), m1, m2, m3);
    ins_min3(__shfl_xor(m2, 16, 32), m1, m2, m3);
    ins_min3(__shfl_xor(m3, 16, 32), m1, m2, m3);

    if (lane < 16) {
        const int e = wave * 16 + lane;
        const long o = (((long)(b * 32 + rb)) * E_DIM + e) * 6;
        cand[o + 0] = a1; cand[o + 1] = a2; cand[o + 2] = a3;
        cand[o + 3] = m1; cand[o + 4] = m2; cand[o + 5] = m3;
    }
}

// ---------------------------------------------------------------------------
// Kernel 2: merge 32 row-block candidates per (b,e), pool, L2-normalize row.
// ---------------------------------------------------------------------------
__global__ __launch_bounds__(128) void weldon_reduce(
    const float* __restrict__ cand,
    float* __restrict__ out)
{
    __shared__ float red[E_DIM];
    const int b = blockIdx.x;
    const int e = threadIdx.x;

    float a1 = NEG_SENT, a2 = NEG_SENT, a3 = NEG_SENT;
    float m1 = POS_SENT, m2 = POS_SENT, m3 = POS_SENT;
#pragma unroll 4
    for (int rb = 0; rb < 32; ++rb) {
        const float* p = cand + (((long)(b * 32 + rb)) * E_DIM + e) * 6;
        ins_max3(p[0], a1, a2, a3);
        ins_max3(p[1], a1, a2, a3);
        ins_max3(p[2], a1, a2, a3);
        ins_min3(p[3], m1, m2, m3);
        ins_min3(p[4], m1, m2, m3);
        ins_min3(p[5], m1, m2, m3);
    }
    const float pooled = (a1 + a2 + a3) + (m1 + m2 + m3);

    red[e] = pooled * pooled;
    __syncthreads();
#pragma unroll
    for (int s = 64; s > 0; s >>= 1) {
        if (e < s) red[e] += red[e + s];
        __syncthreads();
    }
    const float sq = red[0];
    out[b * E_DIM + e] = pooled * rsqrtf(fmaxf(sq, 1e-12f));
}

// ---------------------------------------------------------------------------
extern "C" void kernel_launch(void* const* d_in, const int* in_sizes, int n_in,
                              void* d_out, int out_size, void* d_ws, size_t ws_size,
                              hipStream_t stream)
{
    const float* x = (const float*)d_in[0];   // [32,64,64,512] f32
    const float* W = (const float*)d_in[1];   // [512,128] f32
    float* cand = (float*)d_ws;               // needs 32*32*128*6*4 = 3 MiB
    float* out  = (float*)d_out;              // [32,128] f32

    dim3 grid1(32 /*row-blocks*/, 32 /*batches*/);
    weldon_gemm_pool<<<grid1, dim3(256), 0, stream>>>(x, W, cand);
    weldon_reduce<<<dim3(32), dim3(128), 0, stream>>>(cand, out);
}